// TransformAttention_33981781246629
// MI455X (gfx1250) — compile-verified
//
#include <hip/hip_runtime.h>
#include <hip/hip_bf16.h>

typedef float v2f __attribute__((ext_vector_type(2)));
typedef float v8f __attribute__((ext_vector_type(8)));

#define HID 64
#define B_  16
#define T_  12
#define N_  1024
#define HEADS 8
#define CHUNKS (N_ / HEADS)            // 128
#define MTOT (B_ * T_ * N_)            // 196608 rows per FC GEMM
#define SZ   ((long)MTOT * HID)        // elements per [B,T,N,D] tensor

#if defined(__has_builtin)
# if __has_builtin(__builtin_amdgcn_global_load_async_to_lds_b128)
#  define HAVE_ASYNC_LDS 1
# endif
#endif

#ifdef HAVE_ASYNC_LDS
// Probe-revealed signature: param0 is `global int __vector_size__(16) *`.
typedef int v4i_vs __attribute__((vector_size(16)));
typedef __attribute__((address_space(1))) v4i_vs* g1v4p;
typedef __attribute__((address_space(3))) v4i_vs* l3v4p;
__device__ __forceinline__ void async_cp16(const void* gsrc, void* ldst) {
    __builtin_amdgcn_global_load_async_to_lds_b128((g1v4p)gsrc, (l3v4p)ldst, 0, 0);
}
__device__ __forceinline__ void wait_async0() {
#if __has_builtin(__builtin_amdgcn_s_wait_asynccnt)
    __builtin_amdgcn_s_wait_asynccnt(0);
#else
    asm volatile("s_wait_asynccnt 0x0" ::: "memory");
#endif
}
#endif

// ---------------------------------------------------------------------------
// Zero the BN statistics accumulators (they are atomically accumulated).
// ---------------------------------------------------------------------------
__global__ void zero_stats_kernel(float* __restrict__ p, int n) {
    int i = blockIdx.x * blockDim.x + threadIdx.x;
    if (i < n) p[i] = 0.0f;
}

// ---------------------------------------------------------------------------
// FC GEMM: out[m, e] = sum_d in[m, d] * W[d, e] + b[e]      (M=196608, 64x64)
// W is staged to LDS TRANSPOSED (Wt[e][d]) so every WMMA B-fragment is one
// aligned ds_load_b64. A-fragments are contiguous global b64 loads.
// One wave computes a 16x64 output tile with V_WMMA_F32_16X16X4_F32.
// blockDim = 256 (8 waves), grid = MTOT / (8*16) = 1536.
// Also accumulates per-channel sum / sum^2 for train-mode BatchNorm.
// ---------------------------------------------------------------------------
__global__ __launch_bounds__(256) void fc_gemm_stats(
    const float* __restrict__ in, const float* __restrict__ W,
    const float* __restrict__ b, float* __restrict__ out,
    float* __restrict__ stats /* [128]: sum[64], sumsq[64] */)
{
    __shared__ float Wt[64 * 64];                 // transposed weights
    __shared__ float bl[64];
    __shared__ float sred[128];

    int tid = threadIdx.x;

#ifdef HAVE_ASYNC_LDS
    __shared__ float Wraw[64 * 64];
    // Async DMA 16KB of W into LDS (ASYNCcnt path), then transpose in LDS.
    #pragma unroll
    for (int r = 0; r < 4; ++r) {
        int f4 = tid + r * 256;                   // float4 index 0..1023
        async_cp16(W + f4 * 4, Wraw + f4 * 4);
    }
    wait_async0();
    __syncthreads();
    for (int i = tid; i < 1024; i += 256) {
        float4 w4 = ((const float4*)Wraw)[i];
        int d = (i * 4) >> 6, e = (i * 4) & 63;
        Wt[(e + 0) * 64 + d] = w4.x;
        Wt[(e + 1) * 64 + d] = w4.y;
        Wt[(e + 2) * 64 + d] = w4.z;
        Wt[(e + 3) * 64 + d] = w4.w;
    }
#else
    for (int i = tid; i < 1024; i += 256) {
        float4 w4 = ((const float4*)W)[i];
        int d = (i * 4) >> 6, e = (i * 4) & 63;
        Wt[(e + 0) * 64 + d] = w4.x;
        Wt[(e + 1) * 64 + d] = w4.y;
        Wt[(e + 2) * 64 + d] = w4.z;
        Wt[(e + 3) * 64 + d] = w4.w;
    }
#endif
    if (tid < 64) bl[tid] = b[tid];
    if (tid < 128) sred[tid] = 0.0f;
    __syncthreads();

    int wave = tid >> 5, lane = tid & 31;
    int lm = lane & 15, lh = lane >> 4;
    long row0 = ((long)blockIdx.x * 8 + wave) * 16;

    v8f z = {0.f, 0.f, 0.f, 0.f, 0.f, 0.f, 0.f, 0.f};
    v8f acc[4] = {z, z, z, z};

    const float* arow = in + (row0 + lm) * HID;   // each lane owns one A row
    #pragma unroll
    for (int kk = 0; kk < 16; ++kk) {
        int kb = kk * 4 + 2 * lh;                 // A 16x4 f32 layout: K split by lane half
        v2f a = *(const v2f*)(arow + kb);         // global_load_b64
        #pragma unroll
        for (int n = 0; n < 4; ++n) {
            int col = n * 16 + lm;
            v2f bb = *(const v2f*)&Wt[col * 64 + kb];   // single ds_load_b64
            acc[n] = __builtin_amdgcn_wmma_f32_16x16x4_f32(
                false, a, false, bb, (short)0, acc[n], false, false);
        }
    }

    // Bias, store pre-BN result, reduce per-channel stats.
    #pragma unroll
    for (int n = 0; n < 4; ++n) {
        int col = n * 16 + lm;
        float bias = bl[col];
        float s = 0.f, sq = 0.f;
        #pragma unroll
        for (int r = 0; r < 8; ++r) {
            float v = acc[n][r] + bias;           // C layout: row = row0 + r + 8*lh
            out[(row0 + r + 8 * lh) * HID + col] = v;
            s += v; sq += v * v;
        }
        // lanes L and L^16 hold the same column (different rows): combine.
        s  += __shfl_xor(s, 16, 32);
        sq += __shfl_xor(sq, 16, 32);
        if (lh == 0) {
            atomicAdd(&sred[col], s);             // ds_add_f32
            atomicAdd(&sred[64 + col], sq);
        }
    }
    __syncthreads();
    if (tid < 128) atomicAdd(&stats[tid], sred[tid]);  // one global atomic / chan / block
}

// ---------------------------------------------------------------------------
// BN finalize: scale = gamma * rsqrt(var + eps); shift = beta - mean*scale.
// ---------------------------------------------------------------------------
__global__ void bn_finalize(const float* __restrict__ stats,
                            const float* __restrict__ gamma,
                            const float* __restrict__ beta,
                            float* __restrict__ scsh, float invM)
{
    int c = threadIdx.x;                          // 64 threads
    float mean = stats[c] * invM;
    float var  = stats[64 + c] * invM - mean * mean;
    float sc   = gamma[c] * rsqrtf(var + 1e-5f);
    scsh[c] = sc;
    scsh[64 + c] = beta[c] - mean * sc;
}

// ---------------------------------------------------------------------------
// Fused chunked attention. One wave per (cb, t) block; cb = c*B + b.
//   attn[d1,d2] = softmax_d2( (1/8) * sum_j q[j,d1] k[j,d2] )   (j = 0..7)
//   y[d1,h]     = sum_d2 attn[d1,d2] v[h,d2]
// q and k are staged TRANSPOSED in LDS (qt[d][j], kt[d][j]) so both WMMA
// fragments of the first matmul are single aligned ds_load_b64 ops.
// BN (+ReLU for q,k) applied while staging. Output written directly in the
// reassembled [B,T,N,D] layout: n = c*8 + h.
// ---------------------------------------------------------------------------
__global__ __launch_bounds__(32) void attention_kernel(
    const float* __restrict__ qp, const float* __restrict__ kp,
    const float* __restrict__ vp,
    const float* __restrict__ sq, const float* __restrict__ sk,
    const float* __restrict__ sv,
    float* __restrict__ out)
{
    __shared__ float qt[64 * 8];                  // qt[d][j] = BNReLU(q_chunk[j][d])
    __shared__ float kt[64 * 8];                  // kt[d][j] = BNReLU(k_chunk[j][d])
    __shared__ float vs[8 * 64];                  // vs[h][d] = BN(v_chunk[h][d])
    __shared__ float at[16 * 68];                 // one 16x64 attn tile (+pad)

    int lane = threadIdx.x;
    int blk  = blockIdx.x;
    int t  = blk % T_;
    int cb = blk / T_;
    int c  = cb >> 4;                             // cb = c*16 + b
    int b  = cb & 15;
    long base = (((long)b * T_ + t) * N_ + (long)c * HEADS) * HID;  // 8 contiguous rows

    // Stage q/k/v chunks (512 floats each, contiguous) applying BN (+ReLU).
    #pragma unroll
    for (int i = 0; i < 4; ++i) {
        int idx = (lane + 32 * i);                // float4 index 0..127
        int j = idx >> 4;                         // head row 0..7
        int d = (idx * 4) & 63;
        float4 q4 = ((const float4*)(qp + base))[idx];
        float4 k4 = ((const float4*)(kp + base))[idx];
        float4 v4 = ((const float4*)(vp + base))[idx];
        float4 vr;
        qt[(d + 0) * 8 + j] = fmaxf(sq[d + 0] * q4.x + sq[64 + d + 0], 0.f);
        qt[(d + 1) * 8 + j] = fmaxf(sq[d + 1] * q4.y + sq[64 + d + 1], 0.f);
        qt[(d + 2) * 8 + j] = fmaxf(sq[d + 2] * q4.z + sq[64 + d + 2], 0.f);
        qt[(d + 3) * 8 + j] = fmaxf(sq[d + 3] * q4.w + sq[64 + d + 3], 0.f);
        kt[(d + 0) * 8 + j] = fmaxf(sk[d + 0] * k4.x + sk[64 + d + 0], 0.f);
        kt[(d + 1) * 8 + j] = fmaxf(sk[d + 1] * k4.y + sk[64 + d + 1], 0.f);
        kt[(d + 2) * 8 + j] = fmaxf(sk[d + 2] * k4.z + sk[64 + d + 2], 0.f);
        kt[(d + 3) * 8 + j] = fmaxf(sk[d + 3] * k4.w + sk[64 + d + 3], 0.f);
        vr.x = sv[d + 0] * v4.x + sv[64 + d + 0]; // value FC: no ReLU
        vr.y = sv[d + 1] * v4.y + sv[64 + d + 1];
        vr.z = sv[d + 2] * v4.z + sv[64 + d + 2];
        vr.w = sv[d + 3] * v4.w + sv[64 + d + 3];
        ((float4*)vs)[idx] = vr;
    }
    __syncthreads();

    int lm = lane & 15, lh = lane >> 4;
    v8f z = {0.f, 0.f, 0.f, 0.f, 0.f, 0.f, 0.f, 0.f};
    v2f z2 = {0.f, 0.f};

    long obase = (((long)b * T_ + t) * N_ + (long)c * HEADS + lm) * HID;

    for (int mt = 0; mt < 4; ++mt) {
        int m0 = mt * 16;
        // ---- attn tile: A = q^T (16 x 8), B = k (8 x 16n), 2 k-steps x 4 n-tiles
        v8f accn[4] = {z, z, z, z};
        #pragma unroll
        for (int j0 = 0; j0 < 8; j0 += 4) {
            int jb = j0 + 2 * lh;
            v2f a = *(const v2f*)&qt[(m0 + lm) * 8 + jb];      // ds_load_b64
            #pragma unroll
            for (int n = 0; n < 4; ++n) {
                int col = n * 16 + lm;
                v2f bb = *(const v2f*)&kt[col * 8 + jb];       // ds_load_b64
                accn[n] = __builtin_amdgcn_wmma_f32_16x16x4_f32(
                    false, a, false, bb, (short)0, accn[n], false, false);
            }
        }
        // ---- softmax over the 64 columns of each row (rows live at fixed (r,half))
        #pragma unroll
        for (int r = 0; r < 8; ++r) {
            float v0 = accn[0][r] * 0.125f, v1 = accn[1][r] * 0.125f;
            float v2 = accn[2][r] * 0.125f, v3 = accn[3][r] * 0.125f;
            float mx = fmaxf(fmaxf(v0, v1), fmaxf(v2, v3));
            #pragma unroll
            for (int m = 8; m >= 1; m >>= 1) mx = fmaxf(mx, __shfl_xor(mx, m, 32));
            v0 = __expf(v0 - mx); v1 = __expf(v1 - mx);
            v2 = __expf(v2 - mx); v3 = __expf(v3 - mx);
            float s = v0 + v1 + v2 + v3;
            #pragma unroll
            for (int m = 8; m >= 1; m >>= 1) s += __shfl_xor(s, m, 32);
            float inv = 1.0f / s;
            accn[0][r] = v0 * inv; accn[1][r] = v1 * inv;
            accn[2][r] = v2 * inv; accn[3][r] = v3 * inv;
        }
        // ---- C-layout -> LDS so we can re-fragment as WMMA A operand
        #pragma unroll
        for (int n = 0; n < 4; ++n)
            #pragma unroll
            for (int r = 0; r < 8; ++r)
                at[(r + 8 * lh) * 68 + n * 16 + lm] = accn[n][r];
        __syncthreads();

        // ---- y_tile[16, 8pad16] = at[16 x 64] @ v^T[64 x 8], 16 k-steps
        v8f yac = z;
        #pragma unroll
        for (int kk = 0; kk < 16; ++kk) {
            int kb = kk * 4 + 2 * lh;
            v2f a  = *(const v2f*)&at[lm * 68 + kb];            // ds_load_b64
            v2f tv = *(const v2f*)&vs[(lm & 7) * 64 + kb];      // ds_load_b64
            v2f bb = (lm < 8) ? tv : z2;                        // pad cols 8..15
            yac = __builtin_amdgcn_wmma_f32_16x16x4_f32(
                false, a, false, bb, (short)0, yac, false, false);
        }
        // ---- write out[b,t, c*8+h, d]: d = m0 + 8*lh + r is contiguous in r
        if (lm < HEADS) {
            float4 lo = {yac[0], yac[1], yac[2], yac[3]};
            float4 hi = {yac[4], yac[5], yac[6], yac[7]};
            long o = obase + m0 + 8 * lh;
            *(float4*)(out + o)     = lo;
            *(float4*)(out + o + 4) = hi;
        }
        __syncthreads();                           // 'at' reused next m-tile
    }
}

// ---------------------------------------------------------------------------
// Final BN apply (no ReLU): out = scale[ch]*x + shift[ch], float4 vectorized.
// ---------------------------------------------------------------------------
__global__ __launch_bounds__(256) void bn_apply(
    const float* __restrict__ in, const float* __restrict__ scsh,
    float* __restrict__ out, long n4)
{
    long i = (long)blockIdx.x * blockDim.x + threadIdx.x;
    if (i >= n4) return;
    int d = (int)((i * 4) & 63);
    float4 v = ((const float4*)in)[i];
    float4 r;
    r.x = scsh[d + 0] * v.x + scsh[64 + d + 0];
    r.y = scsh[d + 1] * v.y + scsh[64 + d + 1];
    r.z = scsh[d + 2] * v.z + scsh[64 + d + 2];
    r.w = scsh[d + 3] * v.w + scsh[64 + d + 3];
    ((float4*)out)[i] = r;
}

// ---------------------------------------------------------------------------
extern "C" void kernel_launch(void* const* d_in, const int* in_sizes, int n_in,
                              void* d_out, int out_size, void* d_ws, size_t ws_size,
                              hipStream_t stream) {
    const float* x     = (const float*)d_in[0];
    const float* his   = (const float*)d_in[1];
    const float* pred  = (const float*)d_in[2];
    const float* Wq = (const float*)d_in[3],  *bq = (const float*)d_in[4];
    const float* gq = (const float*)d_in[5],  *Bq = (const float*)d_in[6];
    const float* Wk = (const float*)d_in[7],  *bk = (const float*)d_in[8];
    const float* gk = (const float*)d_in[9],  *Bk = (const float*)d_in[10];
    const float* Wv = (const float*)d_in[11], *bv = (const float*)d_in[12];
    const float* gv = (const float*)d_in[13], *Bv = (const float*)d_in[14];
    const float* Wo = (const float*)d_in[15], *bo = (const float*)d_in[16];
    const float* go = (const float*)d_in[17], *Bo = (const float*)d_in[18];

    float* f = (float*)d_ws;
    float* q_pre = f;                 // [SZ]
    float* k_pre = f + SZ;            // [SZ]
    float* v_pre = f + 2 * SZ;        // [SZ]
    float* attn  = f + 3 * SZ;        // [SZ]
    float* o_pre = q_pre;             // reuse: q_pre dead after attention
    float* stats = f + 4 * SZ;        // [512]: 4 FCs x (sum[64], sumsq[64])
    float* scsh  = stats + 512;       // [512]: 4 FCs x (scale[64], shift[64])

    const float invM = 1.0f / (float)MTOT;
    const int gemmGrid = MTOT / (8 * 16);         // 1536

    zero_stats_kernel<<<2, 256, 0, stream>>>(stats, 512);

    fc_gemm_stats<<<gemmGrid, 256, 0, stream>>>(pred, Wq, bq, q_pre, stats + 0);
    fc_gemm_stats<<<gemmGrid, 256, 0, stream>>>(his,  Wk, bk, k_pre, stats + 128);
    fc_gemm_stats<<<gemmGrid, 256, 0, stream>>>(x,    Wv, bv, v_pre, stats + 256);

    bn_finalize<<<1, 64, 0, stream>>>(stats + 0,   gq, Bq, scsh + 0,   invM);
    bn_finalize<<<1, 64, 0, stream>>>(stats + 128, gk, Bk, scsh + 128, invM);
    bn_finalize<<<1, 64, 0, stream>>>(stats + 256, gv, Bv, scsh + 256, invM);

    attention_kernel<<<CHUNKS * B_ * T_, 32, 0, stream>>>(
        q_pre, k_pre, v_pre, scsh + 0, scsh + 128, scsh + 256, attn);

    fc_gemm_stats<<<gemmGrid, 256, 0, stream>>>(attn, Wo, bo, o_pre, stats + 384);
    bn_finalize<<<1, 64, 0, stream>>>(stats + 384, go, Bo, scsh + 384, invM);
    bn_apply<<<(int)((SZ / 4 + 255) / 256), 256, 0, stream>>>(
        o_pre, scsh + 384, (float*)d_out, SZ / 4);
}